// OptimizedTensorProduct_79912161509904
// MI455X (gfx1250) — compile-verified
//
#include <hip/hip_runtime.h>
#include <hip/hip_bf16.h>
#include <math.h>

#define NTHREADS 256
#define WAVES    8
#define ZT       16      // batch rows per workgroup
#define FEAT     288
#define WBLK     8       // w-block for the t-buffer
#define NPATH    11
#define BATCH    8192
#define TCHUNK   264     // skewed chunk stride (32 lanes * 8 + 8 pad) for t fragments

typedef float v2f __attribute__((ext_vector_type(2)));
typedef float v4f __attribute__((ext_vector_type(4)));
typedef float v8f __attribute__((ext_vector_type(8)));

// path tables: (i1,i2,io) in reference enumeration order; irrep index == l here
__constant__ int c_i1[NPATH]  = {0,0,0,1,1,1,1,2,2,2,2};
__constant__ int c_i2[NPATH]  = {0,1,2,0,1,1,2,0,1,2,2};
__constant__ int c_io[NPATH]  = {0,1,2,1,0,2,1,2,1,0,2};
__constant__ int c_offs[3]    = {0,32,128};   // feature offsets of each irrep block
__constant__ int c_ab[3]      = {0,512,2048}; // A-pack float base per irrep (L*512 each)
__constant__ int c_fan[3]     = {3,4,4};      // #paths feeding each output irrep (fan = cnt*1024)

// ---------------------------------------------------------------------------
// Pre-kernel: generate real-basis Wigner-3j tensors (fp64), e3nn convention,
// with alpha = sqrt((2lo+1)/fan) folded in.  Output: w3tab[p*125 + (a*5+b)*5+c]
// ---------------------------------------------------------------------------
struct cpx { double re, im; };
__device__ inline cpx cmul(cpx a, cpx b){ return {a.re*b.re - a.im*b.im, a.re*b.im + a.im*b.re}; }
__device__ inline cpx conjc(cpx a){ return {a.re, -a.im}; }

__device__ double dfact(int n){ double r = 1.0; for (int i = 2; i <= n; ++i) r *= (double)i; return r; }

__device__ double cg_dev(int j1,int m1,int j2,int m2,int j3,int m3){
  if (m1 + m2 != m3) return 0.0;
  if (j3 < abs(j1-j2) || j3 > j1+j2) return 0.0;
  double pre = (2.0*j3+1.0)*dfact(j3+j1-j2)*dfact(j3-j1+j2)*dfact(j1+j2-j3)/dfact(j1+j2+j3+1);
  pre *= dfact(j3+m3)*dfact(j3-m3)*dfact(j1-m1)*dfact(j1+m1)*dfact(j2-m2)*dfact(j2+m2);
  double s = 0.0;
  for (int k = 0; k <= j1+j2-j3; ++k){
    int a0=k, a1=j1+j2-j3-k, a2=j1-m1-k, a3=j2+m2-k, a4=j3-j2+m1+k, a5=j3-j1-m2+k;
    if (a0<0||a1<0||a2<0||a3<0||a4<0||a5<0) continue;
    double d = dfact(a0)*dfact(a1)*dfact(a2)*dfact(a3)*dfact(a4)*dfact(a5);
    s += ((k & 1) ? -1.0 : 1.0) / d;
  }
  return sqrt(pre) * s;
}

// row a (0..2l) of the complex->real change of basis Q
__device__ void qrow(int l, int a, cpx* row){
  for (int c = 0; c < 5; ++c){ row[c].re = 0.0; row[c].im = 0.0; }
  const double is2 = 0.70710678118654752440;
  if (a == l){ row[l].re = 1.0; return; }
  if (a > l){
    int m = a - l; double s = (m & 1) ? -1.0 : 1.0;
    row[l+m].re = s * is2; row[l-m].re = is2;
  } else {
    int m = l - a; double s = (m & 1) ? -1.0 : 1.0;
    row[l-m].im = is2; row[l+m].im = -s * is2;
  }
}

__global__ void wigner3j_kernel(float* __restrict__ w3tab){
  const int p = blockIdx.x;
  const int tid = threadIdx.x;            // 128 threads, 125 active
  const int l1 = c_i1[p], l2 = c_i2[p], lo = c_io[p];
  const int n1 = 2*l1+1, n2 = 2*l2+1, n3 = 2*lo+1;
  __shared__ double Cr[125], Tr[125], Ti[125];
  __shared__ double s_scale; __shared__ int s_useRe;

  if (tid < 125){
    const int a = tid/25, b = (tid/5)%5, c = tid%5;
    double cr = 0.0;
    if (a < n1 && b < n2 && c < n3){
      const int m1 = a - l1, m2 = b - l2, m3 = c - lo;
      if (m1 + m2 + m3 == 0){
        const double sgn = ((l1 - l2 - m3) & 1) ? -1.0 : 1.0;
        cr = sgn * cg_dev(l1, m1, l2, m2, lo, -m3);
      }
    }
    Cr[tid] = cr;
  }
  __syncthreads();
  if (tid < 125){
    const int a = tid/25, b = (tid/5)%5, c = tid%5;
    double tre = 0.0, tim = 0.0;
    if (a < n1 && b < n2 && c < n3){
      cpx q1[5], q2[5], q3[5];
      qrow(l1, a, q1); qrow(l2, b, q2); qrow(lo, c, q3);
      for (int mm = 0; mm < n1; ++mm){
        cpx f1 = conjc(q1[mm]);
        for (int nn = 0; nn < n2; ++nn){
          cpx f12 = cmul(f1, conjc(q2[nn]));
          for (int pp = 0; pp < n3; ++pp){
            double cv = Cr[(mm*5+nn)*5+pp];
            if (cv == 0.0) continue;
            cpx f = cmul(f12, conjc(q3[pp]));
            tre += cv * f.re; tim += cv * f.im;
          }
        }
      }
    }
    Tr[tid] = tre; Ti[tid] = tim;
  }
  __syncthreads();
  if (tid == 0){
    double sre = 0.0, sim = 0.0;
    for (int k = 0; k < 125; ++k){ sre += fabs(Tr[k]); sim += fabs(Ti[k]); }
    const int useRe = (sre >= sim) ? 1 : 0;
    double nrm = 0.0;
    for (int k = 0; k < 125; ++k){ double v = useRe ? Tr[k] : Ti[k]; nrm += v*v; }
    nrm = sqrt(nrm);
    const double alpha = sqrt((2.0*lo + 1.0) / (1024.0 * (double)c_fan[lo]));
    s_scale = alpha / nrm; s_useRe = useRe;
  }
  __syncthreads();
  if (tid < 125)
    w3tab[p*125 + tid] = (float)((s_useRe ? Tr[tid] : Ti[tid]) * s_scale);
}

// ---------------------------------------------------------------------------
// Main tensor-product kernel.
// Stage 1: t = x1 @ W via v_wmma_f32_16x16x4_f32.
//   A fragments come from a pre-packed LDS image of x1 (8 x ds_load_b64,
//   immediate offsets).  B fragments are loaded from global once per
//   (dw,vtile) job and reused across all L1 'i' chains.  D fragments are
//   stored fragment-native (2 x ds_store_b128) into a skewed t buffer.
// Stage 2: per-(z,w) register contraction with x2 and the Wigner tensor.
// x2 tile is staged global->LDS with async-tensor path
// (global_load_async_to_lds_b128, ASYNCcnt).
// ---------------------------------------------------------------------------
template<int L1, int L2, int Lo>
__device__ void run_path(const float* __restrict__ Wp, const float* __restrict__ w3s,
                         const float* __restrict__ apk, const float* __restrict__ x2s,
                         float* __restrict__ ys, float* __restrict__ tbf,
                         int ab1, int o2, int oo, int tid)
{
  const int lane = tid & 31;
  const int wid  = tid >> 5;
  const int mrow = lane & 15;   // M row of A / N column of B
  const int half = lane >> 4;   // K-pair selector

  for (int w0 = 0; w0 < 32; w0 += WBLK){
    // ---- Stage 1 ----------------------------------------------------------
#pragma unroll
    for (int rep = 0; rep < 2; ++rep){
      const int j   = wid + rep*WAVES;        // 0..15: (dw, vtile)
      const int dw  = j >> 1;
      const int vt  = j & 1;
      const int w   = w0 + dw;
      const int colv = vt*16 + mrow;          // v column
      const float* Bp = Wp + (size_t)colv*32 + (size_t)w;   // + u*1024
      float b0[8], b1[8];
#pragma unroll
      for (int kk = 0; kk < 8; ++kk){         // B loaded once per job
        const int u0 = kk*4 + 2*half;
        b0[kk] = Bp[(size_t)u0*1024];
        b1[kk] = Bp[(size_t)(u0+1)*1024];
      }
      const int chunk0 = (dw*2 + vt)*L1;
#pragma unroll
      for (int i = 0; i < L1; ++i){
        const float* Af = apk + ab1 + i*512 + half*32 + mrow*2;
        v8f acc = {0.f,0.f,0.f,0.f,0.f,0.f,0.f,0.f};
#if defined(__AMDGCN__)
#pragma unroll
        for (int kk = 0; kk < 8; ++kk){       // K = u, 4 per WMMA
          v2f A = *(const v2f*)(Af + kk*64);  // ds_load_b64, immediate offset
          v2f B; B[0] = b0[kk]; B[1] = b1[kk];
          acc = __builtin_amdgcn_wmma_f32_16x16x4_f32(
              false, A, false, B, (short)0, acc, false, false);
        }
#endif
        float* tp = tbf + (chunk0 + i)*TCHUNK + lane*8;
        v4f lo = {acc[0], acc[1], acc[2], acc[3]};
        v4f hi = {acc[4], acc[5], acc[6], acc[7]};
        *(v4f*)tp       = lo;                 // ds_store_b128
        *(v4f*)(tp + 4) = hi;                 // ds_store_b128
      }
    }
    __syncthreads();

    // ---- Stage 2: y[z,w,ko] += sum_{i,v,j} t * x2 * w3 ---------------------
    if (tid < ZT*WBLK){
      const int z  = tid >> 3;
      const int dw = tid & (WBLK-1);
      const int w  = w0 + dw;
      const int zoff = (z >> 3)*128 + (z & 7);   // fragment position of row z
      float pm[L1][L2];
#pragma unroll
      for (int i = 0; i < L1; ++i)
#pragma unroll
        for (int jj = 0; jj < L2; ++jj) pm[i][jj] = 0.f;
#pragma unroll
      for (int vt = 0; vt < 2; ++vt){
        const int cbase = (dw*2 + vt)*L1;
        for (int m = 0; m < 16; ++m){
          const int v = vt*16 + m;
          float xv[L2];
#pragma unroll
          for (int jj = 0; jj < L2; ++jj) xv[jj] = x2s[z*FEAT + o2 + v*L2 + jj];
#pragma unroll
          for (int i = 0; i < L1; ++i){
            const float tv = tbf[(cbase + i)*TCHUNK + zoff + m*8];
#pragma unroll
            for (int jj = 0; jj < L2; ++jj) pm[i][jj] += tv * xv[jj];
          }
        }
      }
#pragma unroll
      for (int ko = 0; ko < Lo; ++ko){
        float a2 = 0.f;
#pragma unroll
        for (int i = 0; i < L1; ++i)
#pragma unroll
          for (int jj = 0; jj < L2; ++jj)
            a2 += w3s[(i*5 + jj)*5 + ko] * pm[i][jj];
        ys[z*FEAT + oo + w*Lo + ko] += a2;
      }
    }
    __syncthreads();
  }
}

__global__ void __launch_bounds__(NTHREADS)
tp_kernel(const float* __restrict__ x1, const float* __restrict__ x2,
          const float* __restrict__ wgt, const float* __restrict__ w3tab,
          float* __restrict__ out)
{
  extern __shared__ float smem[];
  float* apk = smem;                  // 4608: x1 tile in A-fragment layout
  float* x2s = apk + ZT*FEAT;         // 4608 (filled by async-to-LDS)
  float* ys  = x2s + ZT*FEAT;         // 4608
  float* tbf = ys  + ZT*FEAT;         // 80*264 = 21120 (skewed fragment chunks)
  float* w3s = tbf + WBLK*2*5*TCHUNK; // 128

  const int tid = threadIdx.x;
  const size_t z0 = (size_t)blockIdx.x * ZT;

#if defined(__AMDGCN__)
  // ---- async global->LDS staging of the (contiguous) x2 tile --------------
  // Dynamic-LDS region starts at offset 0 (no static __shared__ here), so the
  // LDS byte offset of x2s is ZT*FEAT*4.  GVS mode: SADDR = x2 tile base,
  // VADDR = 32-bit byte offset, VDST = LDS byte address.  Tracked by ASYNCcnt.
  {
    const float* gsrc = x2 + z0*FEAT;
    for (int c = tid; c < (ZT*FEAT)/4; c += NTHREADS){    // 1152 x 16B chunks
      unsigned ldsoff = (unsigned)(ZT*FEAT*sizeof(float)) + (unsigned)c*16u;
      unsigned goff   = (unsigned)c*16u;
      asm volatile("global_load_async_to_lds_b128 %0, %1, %2"
                   :: "v"(ldsoff), "v"(goff), "s"(gsrc) : "memory");
    }
  }
#endif

  // Load x1 tile pre-permuted into WMMA A-fragment order:
  //   apk[ab(irr) + i*512 + kk*64 + h*32 + z*2 + e], u = kk*4 + 2h + e
  for (int idx = tid; idx < ZT*FEAT; idx += NTHREADS){
    const float v1 = x1[z0*FEAT + idx];
    ys[idx] = 0.f;
    const int z = idx / FEAT;
    const int f = idx - z*FEAT;
    int irr, u, i;
    if (f < 32)       { irr = 0; u = f;             i = 0;           }
    else if (f < 128) { irr = 1; u = (f-32)/3;      i = (f-32)%3;    }
    else              { irr = 2; u = (f-128)/5;     i = (f-128)%5;   }
    const int kk = u >> 2, h = (u >> 1) & 1, e = u & 1;
    apk[c_ab[irr] + i*512 + kk*64 + h*32 + z*2 + e] = v1;
  }
#if defined(__AMDGCN__)
  asm volatile("s_wait_asynccnt 0" ::: "memory");   // drain async x2 transfers
#endif
  __syncthreads();

  for (int p = 0; p < NPATH; ++p){
    const int i1 = c_i1[p], i2 = c_i2[p], io = c_io[p];
    const int ab1 = c_ab[i1], o2 = c_offs[i2], oo = c_offs[io];
    const float* Wp = wgt + (size_t)p * 32768;
    if (tid < 125) w3s[tid] = w3tab[p*125 + tid];      // visible after stage-1 barrier
    if (p + 1 < NPATH)                                  // warm L2/L1 for next path
      __builtin_prefetch(wgt + (size_t)(p+1)*32768 + (size_t)tid*128, 0, 1);
    switch (p){
      case 0:  run_path<1,1,1>(Wp,w3s,apk,x2s,ys,tbf,ab1,o2,oo,tid); break;
      case 1:  run_path<1,3,3>(Wp,w3s,apk,x2s,ys,tbf,ab1,o2,oo,tid); break;
      case 2:  run_path<1,5,5>(Wp,w3s,apk,x2s,ys,tbf,ab1,o2,oo,tid); break;
      case 3:  run_path<3,1,3>(Wp,w3s,apk,x2s,ys,tbf,ab1,o2,oo,tid); break;
      case 4:  run_path<3,3,1>(Wp,w3s,apk,x2s,ys,tbf,ab1,o2,oo,tid); break;
      case 5:  run_path<3,3,5>(Wp,w3s,apk,x2s,ys,tbf,ab1,o2,oo,tid); break;
      case 6:  run_path<3,5,3>(Wp,w3s,apk,x2s,ys,tbf,ab1,o2,oo,tid); break;
      case 7:  run_path<5,1,5>(Wp,w3s,apk,x2s,ys,tbf,ab1,o2,oo,tid); break;
      case 8:  run_path<5,3,3>(Wp,w3s,apk,x2s,ys,tbf,ab1,o2,oo,tid); break;
      case 9:  run_path<5,5,1>(Wp,w3s,apk,x2s,ys,tbf,ab1,o2,oo,tid); break;
      case 10: run_path<5,5,5>(Wp,w3s,apk,x2s,ys,tbf,ab1,o2,oo,tid); break;
    }
  }

  for (int idx = tid; idx < ZT*FEAT; idx += NTHREADS)
    out[z0*FEAT + idx] = ys[idx];
}

// ---------------------------------------------------------------------------
extern "C" void kernel_launch(void* const* d_in, const int* in_sizes, int n_in,
                              void* d_out, int out_size, void* d_ws, size_t ws_size,
                              hipStream_t stream)
{
  (void)in_sizes; (void)n_in; (void)out_size; (void)ws_size;
  const float* x1  = (const float*)d_in[0];
  const float* x2  = (const float*)d_in[1];
  const float* wgt = (const float*)d_in[2];
  float* out   = (float*)d_out;
  float* w3tab = (float*)d_ws;          // 11*125 floats of scratch

  const size_t smem_bytes =
      (size_t)(3*ZT*FEAT + WBLK*2*5*TCHUNK + 128) * sizeof(float);  // 140,288 B

  hipError_t err = hipFuncSetAttribute(reinterpret_cast<const void*>(tp_kernel),
                      hipFuncAttributeMaxDynamicSharedMemorySize, (int)smem_bytes);
  (void)err;

  wigner3j_kernel<<<NPATH, 128, 0, stream>>>(w3tab);
  tp_kernel<<<BATCH/ZT, NTHREADS, smem_bytes, stream>>>(x1, x2, wgt, w3tab, out);
}